// AttentionNN_57432302682181
// MI455X (gfx1250) — compile-verified
//
#include <hip/hip_runtime.h>
#include <math.h>

// ---------------------------------------------------------------------------
// AttentionNN fused kernel for MI455X (gfx1250, wave32, WMMA).
//
// Algebra:  a2 = log_softmax(inputsT@attn)@attnT  collapses to
//           a2[f,s'] = (inputsT @ M)[f,s'] - c[s']
//           M  = attn @ attnT                (16x16 per batch)
//           c  = attn @ lse                  (16 per batch)
//           lse[g] = logsumexp_f (inputsT@attn)[f,g]   <-- only heavy part
// The 2048x2048x16 "raw" GEMM is streamed through V_WMMA_F32_16X16X4_F32
// with an online logsumexp; raw is never materialized. inputs_b is staged
// once per block into LDS via GLOBAL_LOAD_ASYNC_TO_LDS_B128 (ASYNCcnt).
// The logsumexp runs in base-2 (B fragments pre-scaled by log2(e)) so the
// epilogue uses bare v_exp_f32 with no scaling multiplies, and the A-fragment
// ds_loads are software-pipelined one f-tile ahead.
// ---------------------------------------------------------------------------

typedef __attribute__((ext_vector_type(2))) float v2f;
typedef __attribute__((ext_vector_type(8))) float v8f;
typedef int v4i __attribute__((vector_size(16)));   // matches builtin pointee

#define SIMS 16
#define NF   2048
#define NB   32
#define GC   256                 // g-columns per block in k1
#define K1_THREADS 256
#define NWAVES (K1_THREADS / 32)
#define GTILES (GC / 16)         // 16 g-tiles per block
#define FTILES (NF / 16)         // 128 f-tiles

#define LOG2E 1.4426950408889634f
#define LN2   0.6931471805599453f

// LDS layout (floats): inputs[16][IN_STRIDE], attn chunk[16][ATT_STRIDE]
#define IN_STRIDE  2064          // 2048 + 16 pad: rows t and t+2 -> disjoint banks
#define ATT_OFF    (16 * IN_STRIDE)
#define ATT_STRIDE 264           // 256 + 8 pad
#define K1_LDS_FLOATS (ATT_OFF + 16 * ATT_STRIDE)
#define K1_LDS_BYTES  (K1_LDS_FLOATS * 4)

// workspace (floats): c[NB][16] then M[NB][16][16]
#define WS_C_OFF 0
#define WS_M_OFF (NB * 16)
#define WS_FLOATS (NB * 16 + NB * 256)

// ---- gfx1250 async-to-LDS path (probe with fallback) ----------------------
#if defined(__has_builtin)
#if __has_builtin(__builtin_amdgcn_global_load_async_to_lds_b128)
#define HAVE_ASYNC_LDS 1
#endif
#if __has_builtin(__builtin_amdgcn_s_wait_asynccnt)
#define HAVE_WAIT_ASYNC 1
#endif
#endif

__device__ __forceinline__ void copy16_to_lds(const float* __restrict__ src,
                                              float* __restrict__ dst) {
#if defined(HAVE_ASYNC_LDS)
    __builtin_amdgcn_global_load_async_to_lds_b128(
        (__attribute__((address_space(1))) v4i*)(void*)src,
        (__attribute__((address_space(3))) v4i*)(void*)dst, 0, 0);
#else
    *(float4*)dst = *(const float4*)src;
#endif
}

__device__ __forceinline__ void wait_async_lds() {
#if defined(HAVE_ASYNC_LDS)
#if defined(HAVE_WAIT_ASYNC)
    __builtin_amdgcn_s_wait_asynccnt(0);
#else
    asm volatile("s_wait_asynccnt 0x0" ::: "memory");
#endif
#endif
}

__global__ __launch_bounds__(256)
void k0_zero(float* __restrict__ ws) {
    int i = blockIdx.x * 256 + threadIdx.x;
    if (i < WS_FLOATS) ws[i] = 0.0f;
}

// ---------------------------------------------------------------------------
// k1: per (batch, g-chunk): stage panels to LDS, WMMA raw tiles + online
//     base-2 logsumexp over f, accumulate partial c[s'] / M[t][s'] atomics.
// ---------------------------------------------------------------------------
__global__ __launch_bounds__(K1_THREADS)
void k1_lse(const float* __restrict__ data, const float* __restrict__ attn,
            float* __restrict__ ws)
{
    extern __shared__ float lds[];

    const int b       = blockIdx.y;
    const int gchunk0 = blockIdx.x * GC;
    const int tid     = threadIdx.x;
    const int wave    = tid >> 5;
    const int lane    = tid & 31;
    const int half    = lane >> 4;   // 0: lanes 0-15, 1: lanes 16-31
    const int l16     = lane & 15;

    const float* __restrict__ inp = data + (size_t)b * SIMS * NF;
    const float* __restrict__ att = attn + (size_t)b * SIMS * NF;

    // ---- stage inputs[16][2048] and attn chunk[16][256] into LDS ----
    for (int idx = tid; idx < 16 * 512; idx += K1_THREADS) {       // 8192 xf4
        const int t = idx >> 9, q = idx & 511;
        copy16_to_lds(inp + (size_t)t * NF + q * 4,
                      &lds[t * IN_STRIDE + q * 4]);
    }
    for (int idx = tid; idx < 16 * 64; idx += K1_THREADS) {        // 1024 xf4
        const int t = idx >> 6, q = idx & 63;
        copy16_to_lds(att + (size_t)t * NF + gchunk0 + q * 4,
                      &lds[ATT_OFF + t * ATT_STRIDE + q * 4]);
    }
    wait_async_lds();
    __syncthreads();

    // ---- partial M[t][s'] over this g-chunk (from LDS) ----
    {
        const int t  = tid >> 4;
        const int sp = tid & 15;
        const float* at = &lds[ATT_OFF + t  * ATT_STRIDE];
        const float* as = &lds[ATT_OFF + sp * ATT_STRIDE];
        float acc = 0.0f;
        for (int g = 0; g < GC; ++g) acc = fmaf(at[g], as[g], acc);
        atomicAdd(&ws[WS_M_OFF + b * 256 + t * 16 + sp], acc);
    }

    // ---- per-wave g-tiles: 2 tiles/wave, each streams 128 f-tiles ----
    for (int gt = wave; gt < GTILES; gt += NWAVES) {
        const int gl0 = gt * 16;            // local g offset within chunk

        // B fragment (4x16 f32 per WMMA, loop-invariant over f), scaled by
        // log2(e) so WMMA emits raw*log2e -> epilogue uses bare v_exp_f32.
        // VGPR0 -> K = 4kk + 2*half, VGPR1 -> K = 4kk + 2*half + 1, N = l16
        v2f bfr[4];
#pragma unroll
        for (int kk = 0; kk < 4; ++kk) {
            const int t0 = kk * 4 + 2 * half;
            bfr[kk].x = lds[ATT_OFF + t0       * ATT_STRIDE + gl0 + l16] * LOG2E;
            bfr[kk].y = lds[ATT_OFF + (t0 + 1) * ATT_STRIDE + gl0 + l16] * LOG2E;
        }

        // software-pipelined A fragments (16x4 f32 layout)
        v2f acur[4], anxt[4];
#pragma unroll
        for (int kk = 0; kk < 4; ++kk) {
            const int t0 = kk * 4 + 2 * half;
            acur[kk].x = lds[t0       * IN_STRIDE + l16];
            acur[kk].y = lds[(t0 + 1) * IN_STRIDE + l16];
        }

        float m = -INFINITY;   // running max (base-2 domain), 8 rows per lane
        float s = 0.0f;        // running sum of 2^(raw2 - m)

        for (int ft = 0; ft < FTILES; ++ft) {
            const int fn = ((ft + 1) & (FTILES - 1)) * 16;   // prefetch tile
#pragma unroll
            for (int kk = 0; kk < 4; ++kk) {
                const int t0 = kk * 4 + 2 * half;
                anxt[kk].x = lds[t0       * IN_STRIDE + fn + l16];
                anxt[kk].y = lds[(t0 + 1) * IN_STRIDE + fn + l16];
            }

            v8f c = {0.f, 0.f, 0.f, 0.f, 0.f, 0.f, 0.f, 0.f};
#pragma unroll
            for (int kk = 0; kk < 4; ++kk) {
                // raw2_tile[16f x 16g] += A[16x4] * (B*log2e)[4x16]
                c = __builtin_amdgcn_wmma_f32_16x16x4_f32(
                        false, acur[kk], false, bfr[kk], (short)0, c, false, false);
            }

            // online base-2 logsumexp over the 8 f-rows this lane holds
            const float tmax = fmaxf(fmaxf(fmaxf(c[0], c[1]), fmaxf(c[2], c[3])),
                                     fmaxf(fmaxf(c[4], c[5]), fmaxf(c[6], c[7])));
            const float mn = fmaxf(m, tmax);
            const float r  = __builtin_amdgcn_exp2f(m - mn);
            const float e0 = __builtin_amdgcn_exp2f(c[0] - mn);
            const float e1 = __builtin_amdgcn_exp2f(c[1] - mn);
            const float e2 = __builtin_amdgcn_exp2f(c[2] - mn);
            const float e3 = __builtin_amdgcn_exp2f(c[3] - mn);
            const float e4 = __builtin_amdgcn_exp2f(c[4] - mn);
            const float e5 = __builtin_amdgcn_exp2f(c[5] - mn);
            const float e6 = __builtin_amdgcn_exp2f(c[6] - mn);
            const float e7 = __builtin_amdgcn_exp2f(c[7] - mn);
            const float t03 = (e0 + e1) + (e2 + e3);
            const float t47 = (e4 + e5) + (e6 + e7);
            s = fmaf(s, r, t03 + t47);
            m = mn;

#pragma unroll
            for (int kk = 0; kk < 4; ++kk) acur[kk] = anxt[kk];
        }

        // combine the two halves of each g-column (lanes l and l^16)
        const float mo = __shfl_xor(m, 16, 32);
        const float so = __shfl_xor(s, 16, 32);
        const float mc = fmaxf(m, mo);
        const float sc = s  * __builtin_amdgcn_exp2f(m  - mc) +
                         so * __builtin_amdgcn_exp2f(mo - mc);
        // back to natural log: lse = ln2 * (m2 + log2(s2))
        const float lse = LN2 * (mc + __builtin_amdgcn_logf(sc));

        // partial c[s'] = sum_{g in tile} lse[g] * att[s', g] ; s' = l16
        float acc = 0.0f;
#pragma unroll
        for (int gg = 0; gg < 16; ++gg) {
            const float lv = __shfl(lse, gg, 32);   // lse of g0+gg
            acc = fmaf(lv, lds[ATT_OFF + l16 * ATT_STRIDE + gl0 + gg], acc);
        }
        if (lane < 16)
            atomicAdd(&ws[WS_C_OFF + b * 16 + l16], acc);
    }
}

// ---------------------------------------------------------------------------
// k2: per (b, f): a2 = inputsT*M - c ; gate = sigmoid([a2,inputs] @ WT + b);
//     out[(r*32 + b), f] = gate[r] * data[(r*32 + b), f]   (AMP = 1.0)
// ---------------------------------------------------------------------------
#define K2_THREADS 256
__global__ __launch_bounds__(K2_THREADS)
void k2_gate(const float* __restrict__ data, const float* __restrict__ W,
             const float* __restrict__ bias, const float* __restrict__ ws,
             float* __restrict__ out)
{
    __shared__ float sM[256];
    __shared__ float sW[512];
    __shared__ float sc[16];
    __shared__ float sb[16];

    const int b   = blockIdx.y;
    const int tid = threadIdx.x;
    const int f   = blockIdx.x * K2_THREADS + tid;

    sM[tid]       = ws[WS_M_OFF + b * 256 + tid];
    sW[tid]       = W[tid];
    sW[tid + 256] = W[tid + 256];
    if (tid < 16) {
        sc[tid] = ws[WS_C_OFF + b * 16 + tid];
        sb[tid] = bias[tid];
    }
    __syncthreads();

    const float* __restrict__ inp = data + (size_t)b * SIMS * NF;

    float inv[16];
#pragma unroll
    for (int t = 0; t < 16; ++t) inv[t] = inp[(size_t)t * NF + f];

    float a2[16];
#pragma unroll
    for (int sp = 0; sp < 16; ++sp) {
        float acc = -sc[sp];
#pragma unroll
        for (int t = 0; t < 16; ++t) acc = fmaf(inv[t], sM[t * 16 + sp], acc);
        a2[sp] = acc;
    }

#pragma unroll
    for (int r = 0; r < 16; ++r) {
        float acc = sb[r];
#pragma unroll
        for (int k = 0; k < 16; ++k) acc = fmaf(a2[k],  sW[r * 32 + k],      acc);
#pragma unroll
        for (int k = 0; k < 16; ++k) acc = fmaf(inv[k], sW[r * 32 + 16 + k], acc);
        const float g = 1.0f / (1.0f + __expf(-acc));
        const size_t j = (size_t)(r * 32 + b) * NF + f;
        out[j] = g * data[j];   // attention_amplifier == 1.0
    }
}

// ---------------------------------------------------------------------------
extern "C" void kernel_launch(void* const* d_in, const int* in_sizes, int n_in,
                              void* d_out, int out_size, void* d_ws, size_t ws_size,
                              hipStream_t stream)
{
    const float* data = (const float*)d_in[0];   // [512, 2048] f32
    const float* attn = (const float*)d_in[1];   // [512, 2048] f32
    const float* W    = (const float*)d_in[2];   // [16, 32]    f32
    const float* bias = (const float*)d_in[3];   // [16]        f32
    float*       out  = (float*)d_out;           // [512, 2048] f32
    float*       ws   = (float*)d_ws;

    hipLaunchKernelGGL(k0_zero, dim3((WS_FLOATS + 255) / 256), dim3(256), 0, stream, ws);
    hipLaunchKernelGGL(k1_lse,  dim3(NF / GC, NB), dim3(K1_THREADS), K1_LDS_BYTES, stream,
                       data, attn, ws);
    hipLaunchKernelGGL(k2_gate, dim3(NF / K2_THREADS, NB), dim3(K2_THREADS), 0, stream,
                       data, W, bias, ws, out);
}